// KSparse_88811333746941
// MI455X (gfx1250) — compile-verified
//
#include <hip/hip_runtime.h>

#define N      4096
#define KP1    65      // K_SPARSE + 1: rank (from top) of the threshold element
#define TPB    256
#define NWAVES (TPB / 32)

// Order-preserving float-bits <-> u32 key (larger key <=> larger float)
__device__ __forceinline__ unsigned f2key(unsigned u) {
    return u ^ ((unsigned)((int)u >> 31) | 0x80000000u);
}
__device__ __forceinline__ unsigned key2f(unsigned k) {
    return k ^ ((unsigned)((int)(~k) >> 31) | 0x80000000u);
}

// Wave 0 only (tid < 32). Finds the largest bin b such that the suffix count
// sum_{b' >= b} hist[b'] >= rem.  Writes sel[0] = b, sel[1] = rem - (count above b).
// No barriers: pure shfl-based, ~40 instructions.
__device__ __forceinline__ void waveSelect(const unsigned* hist, unsigned rem, unsigned* sel) {
    const int lane = (int)threadIdx.x;          // < 32, lane owns bins [lane*8, lane*8+7]
    unsigned h[8];
    #pragma unroll
    for (int j = 0; j < 8; ++j) h[j] = hist[lane * 8 + j];
    unsigned lsum = 0;
    #pragma unroll
    for (int j = 0; j < 8; ++j) lsum += h[j];
    // inclusive suffix-sum across lanes (this lane + all higher lanes)
    unsigned s = lsum;
    #pragma unroll
    for (int off = 1; off < 32; off <<= 1) {
        unsigned t = __shfl_down(s, (unsigned)off, 32);
        if (lane + off < 32) s += t;
    }
    const unsigned above = s - lsum;            // total count in lanes > lane
    int bestj = -1; unsigned sumAbove = 0;
    unsigned acc = above;
    #pragma unroll
    for (int j = 7; j >= 0; --j) {
        acc += h[j];                            // acc = suffix count at bin lane*8+j
        if (bestj < 0 && acc >= rem) { bestj = j; sumAbove = acc - h[j]; }
    }
    int bestb = (bestj >= 0) ? (lane * 8 + bestj) : -1;
    // max-reduce (bestb, sumAbove) across the wave
    #pragma unroll
    for (int off = 16; off >= 1; off >>= 1) {
        int      ob = __shfl_xor(bestb, off, 32);
        unsigned os = __shfl_xor(sumAbove, off, 32);
        if (ob > bestb) { bestb = ob; sumAbove = os; }
    }
    if (lane == 0) { sel[0] = (unsigned)bestb; sel[1] = rem - sumAbove; }
}

__global__ __launch_bounds__(TPB)
void KSparse_kernel(const float* __restrict__ x, float* __restrict__ out) {
    __shared__ __align__(16) unsigned rowbits[N];       // async landing buffer (16 KB)
    __shared__ unsigned whist[NWAVES][256];             // per-wave pass-0 histograms (8 KB)
    __shared__ unsigned hist[256];                      // reduced / refinement histogram (1 KB)
    __shared__ unsigned sel[2];                         // [0]=chosen byte, [1]=residual rank

    const int tid  = (int)threadIdx.x;
    const int wave = tid >> 5;
    const unsigned long long rowBase = (unsigned long long)blockIdx.x * (unsigned long long)N;
    const float* src = x + rowBase;

    // ---- CDNA5 async copy: global row -> LDS (ASYNCcnt path) ----
    {
        const unsigned ldsBase = (unsigned)(size_t)(&rowbits[0]);
        #pragma unroll
        for (int j = 0; j < 4; ++j) {
            unsigned byteOff = (unsigned)((j * TPB + tid) * 16);   // 16 B per lane
            unsigned ldsAddr = ldsBase + byteOff;
            asm volatile("global_load_async_to_lds_b128 %0, %1, %2"
                         :: "v"(ldsAddr), "v"(byteOff), "s"(src)
                         : "memory");
        }
    }
    #pragma unroll
    for (int w = 0; w < NWAVES; ++w) whist[w][tid] = 0;            // overlap with async copy
    asm volatile("s_wait_asynccnt 0" ::: "memory");
    __syncthreads();

    // ---- Load this thread's 16 elements once (b128), keep keys in registers ----
    unsigned key[16];
    {
        const uint4* rv = reinterpret_cast<const uint4*>(rowbits);
        #pragma unroll
        for (int j = 0; j < 4; ++j) {
            uint4 v = rv[j * TPB + tid];
            key[j * 4 + 0] = f2key(v.x);
            key[j * 4 + 1] = f2key(v.y);
            key[j * 4 + 2] = f2key(v.z);
            key[j * 4 + 3] = f2key(v.w);
        }
    }

    // ---- Pass 0: per-wave histograms of the top key byte ----
    #pragma unroll
    for (int e = 0; e < 16; ++e)
        atomicAdd(&whist[wave][key[e] >> 24], 1u);
    __syncthreads();
    {
        unsigned tot = 0;
        #pragma unroll
        for (int w = 0; w < NWAVES; ++w) tot += whist[w][tid];
        hist[tid] = tot;
    }
    __syncthreads();
    if (tid < 32) waveSelect(hist, KP1, sel);
    __syncthreads();

    unsigned prefix = sel[0] << 24;
    unsigned pmask  = 0xFF000000u;

    // ---- Refine lower 3 bytes: only prefix-matching elements (few) touch atomics ----
    for (int shift = 16; shift >= 0; shift -= 8) {
        const unsigned rem = sel[1];
        hist[tid] = 0;
        __syncthreads();
        #pragma unroll
        for (int e = 0; e < 16; ++e) {
            unsigned k = key[e];
            if ((k & pmask) == prefix) atomicAdd(&hist[(k >> shift) & 255u], 1u);
        }
        __syncthreads();
        if (tid < 32) waveSelect(hist, rem, sel);
        __syncthreads();
        prefix |= sel[0] << shift;
        pmask  |= 0xFFu << shift;
    }
    const unsigned thrKey = prefix;   // key of the 65th-largest element

    // ---- Output: keep strictly-greater elements, reconstruct floats from keys ----
    {
        float4* outv = reinterpret_cast<float4*>(out + rowBase);
        #pragma unroll
        for (int j = 0; j < 4; ++j) {
            const unsigned k0 = key[j * 4 + 0], k1 = key[j * 4 + 1];
            const unsigned k2 = key[j * 4 + 2], k3 = key[j * 4 + 3];
            float4 v;
            v.x = (k0 > thrKey) ? __uint_as_float(key2f(k0)) : 0.0f;
            v.y = (k1 > thrKey) ? __uint_as_float(key2f(k1)) : 0.0f;
            v.z = (k2 > thrKey) ? __uint_as_float(key2f(k2)) : 0.0f;
            v.w = (k3 > thrKey) ? __uint_as_float(key2f(k3)) : 0.0f;
            outv[j * TPB + tid] = v;
        }
    }
}

extern "C" void kernel_launch(void* const* d_in, const int* in_sizes, int n_in,
                              void* d_out, int out_size, void* d_ws, size_t ws_size,
                              hipStream_t stream) {
    const float* x = (const float*)d_in[0];
    float* out = (float*)d_out;
    const int rows = in_sizes[0] / N;   // 16384
    KSparse_kernel<<<dim3((unsigned)rows), dim3(TPB), 0, stream>>>(x, out);
}